// pmMSELossFunc_56951266345786
// MI455X (gfx1250) — compile-verified
//
#include <hip/hip_runtime.h>
#include <hip/hip_bf16.h>

typedef __attribute__((ext_vector_type(2))) float v2f;
typedef __attribute__((ext_vector_type(4))) float v4f;
typedef __attribute__((ext_vector_type(8))) float v8f;

#define MAXLEN_W 75825.0f          // (270*270+480*480)/4, exactly representable in fp32
#define CHUNK_ELEMS 69632          // 17*4096 floats share one channel weight
#define BLOCKS_PER_CHUNK 17        // 69632/4 float4 / 17 blocks = 1024 float4 per block
#define THREADS 256                // 8 waves (wave32)

// Full 32-lane wave reduction using V_WMMA_F32_16X16X4_F32.
// Step 1: A[m][k] holds lane partials (lanes 0-15 -> K=0, lanes 16-31 -> K=2,
//         second A-VGPR zeroed), B = ones  =>  D1[m][n] = s_m + s_{m+16} = t_m.
//         D1 VGPR r holds t_r (lanes 0-15) and t_{r+8} (lanes 16-31).
// Step 2: A = ones, B = consecutive D1 VGPR pairs (4x16 layout gives
//         B[0..3][n] = {t_r, t_{r+1}, t_{r+8}, t_{r+9}}), chained through C.
//         After 4 accumulating WMMAs every element of acc = sum of all 32 lanes.
__device__ __forceinline__ float wmma_wave_reduce(float s) {
    v2f a;    a.x = s;    a.y = 0.0f;
    v2f ones; ones.x = 1.0f; ones.y = 1.0f;
    v8f zero = {};
    v8f d1 = __builtin_amdgcn_wmma_f32_16x16x4_f32(
        false, a, false, ones, (short)0, zero, false, false);

    v8f acc = {};
    v2f b;
    b.x = d1.s0; b.y = d1.s1;
    acc = __builtin_amdgcn_wmma_f32_16x16x4_f32(
        false, ones, false, b, (short)0, acc, false, false);
    b.x = d1.s2; b.y = d1.s3;
    acc = __builtin_amdgcn_wmma_f32_16x16x4_f32(
        false, ones, false, b, (short)0, acc, false, false);
    b.x = d1.s4; b.y = d1.s5;
    acc = __builtin_amdgcn_wmma_f32_16x16x4_f32(
        false, ones, false, b, (short)0, acc, false, false);
    b.x = d1.s6; b.y = d1.s7;
    acc = __builtin_amdgcn_wmma_f32_16x16x4_f32(
        false, ones, false, b, (short)0, acc, false, false);
    return acc.s0;   // total replicated in every lane
}

// Stage 1: each block owns 1024 float4s of one channel-chunk; weight is a
// per-block scalar. Streaming non-temporal b128 loads; exact tiling (no
// divergence) so EXEC is all-ones at the WMMA reduction.
__global__ void __launch_bounds__(THREADS)
pm_mse_partial_kernel(const float* __restrict__ predict,
                      const float* __restrict__ target,
                      float* __restrict__ block_partials) {
    const int chunk = blockIdx.x / BLOCKS_PER_CHUNK;   // (bs*3 + ch)
    const int sub   = blockIdx.x % BLOCKS_PER_CHUNK;
    const float w   = ((chunk % 3) == 2) ? MAXLEN_W : 1.0f;

    const long base4 = (long)chunk * (CHUNK_ELEMS / 4) + (long)sub * 1024;
    const v4f* __restrict__ p4 = (const v4f*)predict + base4;
    const v4f* __restrict__ t4 = (const v4f*)target  + base4;

    const int tid = threadIdx.x;
    float s = 0.0f;
#pragma unroll
    for (int i = 0; i < 4; ++i) {
        const int idx = tid + i * THREADS;
        v4f p = __builtin_nontemporal_load(p4 + idx);
        v4f t = __builtin_nontemporal_load(t4 + idx);
        v4f d = t - p;
        s = fmaf(d.x, d.x, s);
        s = fmaf(d.y, d.y, s);
        s = fmaf(d.z, d.z, s);
        s = fmaf(d.w, d.w, s);
    }
    s *= w;

    const float wsum = wmma_wave_reduce(s);   // EXEC all-ones here

    __shared__ float lds[THREADS / 32];
    const int lane = tid & 31;
    const int wave = tid >> 5;
    if (lane == 0) lds[wave] = wsum;
    __syncthreads();
    if (tid == 0) {
        float b = 0.0f;
#pragma unroll
        for (int i = 0; i < THREADS / 32; ++i) b += lds[i];
        block_partials[blockIdx.x] = b;
    }
}

// Stage 2: one block deterministically folds the 3264 block partials and
// applies the 1/(bs*actionlen*seqlen) scale. WMMA again for the wave fold.
__global__ void __launch_bounds__(THREADS)
pm_mse_final_kernel(const float* __restrict__ block_partials, int n,
                    float* __restrict__ out, float inv_denom) {
    const int tid = threadIdx.x;
    float s = 0.0f;
    for (int i = tid; i < n; i += THREADS) s += block_partials[i];
    // loop reconverges; EXEC all-ones again before WMMA
    const float wsum = wmma_wave_reduce(s);

    __shared__ float lds[THREADS / 32];
    if ((tid & 31) == 0) lds[tid >> 5] = wsum;
    __syncthreads();
    if (tid == 0) {
        float t = 0.0f;
#pragma unroll
        for (int i = 0; i < THREADS / 32; ++i) t += lds[i];
        out[0] = t * inv_denom;
    }
}

extern "C" void kernel_launch(void* const* d_in, const int* in_sizes, int n_in,
                              void* d_out, int out_size, void* d_ws, size_t ws_size,
                              hipStream_t stream) {
    const float* predict = (const float*)d_in[0];
    const float* target  = (const float*)d_in[1];
    float* out = (float*)d_out;
    float* ws  = (float*)d_ws;

    const int n_elems  = in_sizes[0];                 // 64*3*17*4096 = 13,369,344
    const int n_chunks = n_elems / CHUNK_ELEMS;       // 192
    const int n_blocks = n_chunks * BLOCKS_PER_CHUNK; // 3264 (13 KB of ws)

    pm_mse_partial_kernel<<<n_blocks, THREADS, 0, stream>>>(predict, target, ws);

    const float inv_denom = 1.0f / (64.0f * 17.0f * 4096.0f); // bs*actionlen*seqlen
    pm_mse_final_kernel<<<1, THREADS, 0, stream>>>(ws, n_blocks, out, inv_denom);
}